// ASA_83133386981682
// MI455X (gfx1250) — compile-verified
//
#include <hip/hip_runtime.h>
#include <hip/hip_bf16.h>

typedef float v2f __attribute__((ext_vector_type(2)));
typedef float v4f __attribute__((ext_vector_type(4)));
typedef float v8f __attribute__((ext_vector_type(8)));

#define B_    32
#define C_    256
#define H_    64
#define W_    64
#define HW_   (H_ * W_)        // 4096
#define CHW_  (C_ * HW_)       // 1<<20
#define NPIX_ (B_ * HW_)       // 131072

// ---------------------------------------------------------------------------
// Kernel 1: per-pixel mean & max over 256 channels.
// Each thread owns 4 consecutive pixels (one float4 column group) and walks
// the channel dimension (stride HW_ floats = 16KB). A wave reads 512B
// contiguous per channel step -> fully coalesced b128 traffic. 2x unroll for
// memory-level parallelism. This pass is the HBM bottleneck (~134MB read);
// default (RT) cache policy deliberately primes L2 with x for pass 3.
// ---------------------------------------------------------------------------
__global__ __launch_bounds__(256) void asa_reduce_kernel(
    const float* __restrict__ x,
    float* __restrict__ s_mean,       // [B, HW]
    float* __restrict__ s_max)        // [B, HW]
{
    const int t = blockIdx.x * 256 + threadIdx.x;       // 0 .. 32767
    const int b = t >> 10;                              // 1024 float4 groups/plane
    const int g = t & 1023;

    const v4f* p = reinterpret_cast<const v4f*>(x + ((size_t)b * CHW_)) + g;
    const size_t cstride = HW_ / 4;                     // 1024 float4 per channel

    v4f s0 = {0.f, 0.f, 0.f, 0.f};
    v4f s1 = {0.f, 0.f, 0.f, 0.f};
    const float NEG = -3.402823466e38f;
    v4f m0 = {NEG, NEG, NEG, NEG};
    v4f m1 = {NEG, NEG, NEG, NEG};

    for (int c = 0; c < C_; c += 2) {
        v4f va = p[(size_t)c * cstride];
        v4f vb = p[(size_t)(c + 1) * cstride];
        s0 += va;
        s1 += vb;
        m0.x = fmaxf(m0.x, va.x); m0.y = fmaxf(m0.y, va.y);
        m0.z = fmaxf(m0.z, va.z); m0.w = fmaxf(m0.w, va.w);
        m1.x = fmaxf(m1.x, vb.x); m1.y = fmaxf(m1.y, vb.y);
        m1.z = fmaxf(m1.z, vb.z); m1.w = fmaxf(m1.w, vb.w);
    }

    const float inv = 1.0f / (float)C_;
    v4f mean4 = (s0 + s1) * inv;
    v4f max4;
    max4.x = fmaxf(m0.x, m1.x); max4.y = fmaxf(m0.y, m1.y);
    max4.z = fmaxf(m0.z, m1.z); max4.w = fmaxf(m0.w, m1.w);

    reinterpret_cast<v4f*>(s_mean)[t] = mean4;
    reinterpret_cast<v4f*>(s_max)[t]  = max4;
}

// Fast sigmoid: v_exp_f32 + v_rcp_f32 (~1 ULP) instead of the IEEE
// div_scale/div_fmas/div_fixup sequence the plain divide lowers to.
__device__ __forceinline__ float asa_sigmoid(float v) {
    return __builtin_amdgcn_rcpf(1.0f + __expf(-v));
}

// ---------------------------------------------------------------------------
// Kernel 2: conv gates + WMMA 1x1 fuse.
// Block = 256 threads = 8 waves; tile = 16(x) x 8(y) pixels, wave w owns row w.
// LDS holds the 2-channel (mean,max) tile with halo 6 (28x20), zero-padded.
// Within a wave: lanes 0-15 compute sigmoid(conv7),sigmoid(conv5) of pixel m,
// lanes 16-31 compute sigmoid(conv9),sigmoid(dilated conv7) of the same pixel
// -- exactly the A-matrix layout of V_WMMA_F32_16X16X4_F32 (16 pixels x K=4
// gates). B is wf[k] broadcast across N. D[m][*] = fused 1x1 conv at pixel m
// in fp32, matching the reference precision.
// ---------------------------------------------------------------------------
__global__ __launch_bounds__(256) void asa_gates_kernel(
    const float* __restrict__ s_mean,
    const float* __restrict__ s_max,
    const float* __restrict__ w7,     // [2,7,7]
    const float* __restrict__ w5,     // [2,5,5]
    const float* __restrict__ w9,     // [2,9,9]
    const float* __restrict__ wd7,    // [2,7,7] dilation 2
    const float* __restrict__ wf,     // [4]
    float* __restrict__ gate)         // [B, HW]
{
    __shared__ float tile[2][20][28];   // halo 6 on each side of 16x8

    const int tid = threadIdx.x;
    const int b  = blockIdx.z;
    const int x0 = blockIdx.x * 16;
    const int y0 = blockIdx.y * 8;

    // Stage s tile (zero padding outside the image).
    for (int i = tid; i < 2 * 20 * 28; i += 256) {
        const int ch = i / (20 * 28);
        const int r  = i % (20 * 28);
        const int yy = r / 28;
        const int xx = r % 28;
        const int gy = y0 + yy - 6;
        const int gx = x0 + xx - 6;
        float v = 0.0f;
        if (gy >= 0 && gy < H_ && gx >= 0 && gx < W_) {
            const float* plane = (ch == 0) ? s_mean : s_max;
            v = plane[b * HW_ + gy * W_ + gx];
        }
        tile[ch][yy][xx] = v;
    }
    __syncthreads();

    const int wave = tid >> 5;
    const int lane = tid & 31;
    const int m    = lane & 15;       // pixel (x offset) within wave's row
    const int hv   = lane >> 4;       // 0: gates {1,2}; 1: gates {3,4}
    const int py   = wave + 6;        // tile-space row of pixel center
    const int px   = m + 6;           // tile-space col of pixel center

    float a0, a1;
    if (hv == 0) {
        float acc7 = 0.f, acc5 = 0.f;
        #pragma unroll
        for (int ch = 0; ch < 2; ++ch) {
            #pragma unroll
            for (int dy = -3; dy <= 3; ++dy)
                #pragma unroll
                for (int dx = -3; dx <= 3; ++dx)
                    acc7 += w7[ch * 49 + (dy + 3) * 7 + (dx + 3)] *
                            tile[ch][py + dy][px + dx];
            #pragma unroll
            for (int dy = -2; dy <= 2; ++dy)
                #pragma unroll
                for (int dx = -2; dx <= 2; ++dx)
                    acc5 += w5[ch * 25 + (dy + 2) * 5 + (dx + 2)] *
                            tile[ch][py + dy][px + dx];
        }
        a0 = asa_sigmoid(acc7);
        a1 = asa_sigmoid(acc5);
    } else {
        float acc9 = 0.f, accd = 0.f;
        #pragma unroll
        for (int ch = 0; ch < 2; ++ch) {
            #pragma unroll
            for (int dy = -4; dy <= 4; ++dy)
                #pragma unroll
                for (int dx = -4; dx <= 4; ++dx)
                    acc9 += w9[ch * 81 + (dy + 4) * 9 + (dx + 4)] *
                            tile[ch][py + dy][px + dx];
            #pragma unroll
            for (int i = 0; i < 7; ++i)          // dilated 7x7, offsets 2i-6
                #pragma unroll
                for (int j = 0; j < 7; ++j)
                    accd += wd7[ch * 49 + i * 7 + j] *
                            tile[ch][py + 2 * i - 6][px + 2 * j - 6];
        }
        a0 = asa_sigmoid(acc9);
        a1 = asa_sigmoid(accd);
    }

    // A: 16x4 fp32 — lanes 0-15: K0(v0),K1(v1); lanes 16-31: K2(v0),K3(v1).
    v2f A; A.x = a0; A.y = a1;
    // B: 4x16 fp32, B[k][n] = wf[k] — same lane/K striping as A.
    v2f Bm;
    Bm.x = hv ? wf[2] : wf[0];
    Bm.y = hv ? wf[3] : wf[1];
    v8f Cz = {};

    // D[m][n] = sum_k sigmoid(gate_k[m]) * wf[k], replicated across n.
    v8f D = __builtin_amdgcn_wmma_f32_16x16x4_f32(
        /*neg_a=*/false, A, /*neg_b=*/false, Bm,
        /*c_mod=*/(short)0, Cz, /*reuse_a=*/false, /*reuse_b=*/false);

    // D row r lives in VGPR r for lanes 0-15 (M=r) and 16-31 (M=r+8):
    // lane 0 writes pixels 0..7, lane 16 writes pixels 8..15.
    if (m == 0) {
        float* gp = gate + b * HW_ + (y0 + wave) * W_ + x0 + hv * 8;
        #pragma unroll
        for (int r = 0; r < 8; ++r)
            gp[r] = asa_sigmoid(D[r]);
    }
}

// ---------------------------------------------------------------------------
// Kernel 3: out = x * gate (broadcast over channels). float4 streaming.
// x reads use default RT policy (they should hit the 192MB L2, primed by
// kernel 1); out stores are non-temporal so the 128MB output stream does not
// evict x lines that haven't been consumed yet.
// ---------------------------------------------------------------------------
__global__ __launch_bounds__(256) void asa_scale_kernel(
    const float* __restrict__ x,
    const float* __restrict__ gate,
    float* __restrict__ out)
{
    const size_t i4 = (size_t)blockIdx.x * 256 + threadIdx.x;
    const size_t i  = i4 * 4;
    const int b   = (int)(i >> 20);           // CHW_ = 2^20
    const int pix = (int)(i & (HW_ - 1));

    const v4f xv = reinterpret_cast<const v4f*>(x)[i4];
    const v4f gv = reinterpret_cast<const v4f*>(gate)[(size_t)b * (HW_ / 4) + (pix >> 2)];

    const v4f o = xv * gv;
    __builtin_nontemporal_store(o, reinterpret_cast<v4f*>(out) + i4);
}

extern "C" void kernel_launch(void* const* d_in, const int* in_sizes, int n_in,
                              void* d_out, int out_size, void* d_ws, size_t ws_size,
                              hipStream_t stream)
{
    const float* x   = (const float*)d_in[0];
    const float* w7  = (const float*)d_in[1];
    const float* w5  = (const float*)d_in[2];
    const float* w9  = (const float*)d_in[3];
    const float* wd7 = (const float*)d_in[4];
    const float* wf  = (const float*)d_in[5];
    float* out = (float*)d_out;

    float* s_mean = (float*)d_ws;                 // NPIX_ floats
    float* s_max  = s_mean + NPIX_;               // NPIX_ floats
    float* gate   = s_max + NPIX_;                // NPIX_ floats  (1.5 MiB total)

    // Pass 1: channel mean/max reduce. 32768 threads, 4 pixels each.
    asa_reduce_kernel<<<dim3(NPIX_ / 4 / 256), dim3(256), 0, stream>>>(x, s_mean, s_max);

    // Pass 2: conv gates + WMMA 1x1 fuse. Tiles 16x8 over 64x64, per batch.
    asa_gates_kernel<<<dim3(W_ / 16, H_ / 8, B_), dim3(256), 0, stream>>>(
        s_mean, s_max, w7, w5, w9, wd7, wf, gate);

    // Pass 3: broadcast multiply. 8M float4 elements.
    asa_scale_kernel<<<dim3((B_ * CHW_ / 4) / 256), dim3(256), 0, stream>>>(x, gate, out);
}